// GRUDecoder_10136122818764
// MI455X (gfx1250) — compile-verified
//
#include <hip/hip_runtime.h>
#include <hip/hip_bf16.h>
#include <math.h>

// GRU decoder step, MI455X (gfx1250, wave32).
// logprobs = log_softmax(W_out @ gru2(relu(emb[tok])) + b_out); hidden_new.
// Entirely HBM-bandwidth bound (~256 MB f32 weights, ~11 us floor at 23.3 TB/s).
// Mat-vecs implemented with V_WMMA_F32_16X16X4_F32: one wave = 16 output rows.

typedef __attribute__((ext_vector_type(2))) float v2f;
typedef __attribute__((ext_vector_type(8))) float v8f;

#define GRU_H 1024
#define GRU_L 2
#define GRU_V 50257

// One wave accumulates 16 consecutive rows of y = W[row..row+15] . v over K range
// [k0,k1). Returns the 16x16 f32 WMMA accumulator (B broadcast across columns, so
// lane 0 holds y[row+0..7] in acc[0..7], lane 16 holds y[row+8..15]).
// EXEC must be all 1s when called (no divergence before/inside the loop).
__device__ __forceinline__ v8f wmma_row16_dot(const float* __restrict__ W, int ldw,
                                              const float* __restrict__ vs,
                                              int row, int k0, int k1) {
  const int lane = (int)(threadIdx.x & 31u);
  const int khalf = (lane >> 4) << 1;                 // 0 for lanes 0-15, 2 for 16-31
  const float* wp = W + (size_t)row * (size_t)ldw + khalf;
  const float* vp = vs + khalf;
  v8f acc = {};
#pragma unroll 8
  for (int k = k0; k < k1; k += 4) {
    // A 16x4 f32 fragment: VGPR0 = K0(l0-15)/K2(l16-31), VGPR1 = K1/K3 -> float2 load.
    v2f a = *(const v2f*)(wp + k);
    // B 4x16 f32 fragment, broadcast across N: VGPR0 = rows K0/K2, VGPR1 = K1/K3.
    v2f b = *(const v2f*)(vp + k);
    acc = __builtin_amdgcn_wmma_f32_16x16x4_f32(false, a, false, b, (short)0, acc,
                                                false, false);
  }
  return acc;
}

// x = relu(emb[tok])
__global__ __launch_bounds__(256) void embed_relu_kernel(const int* __restrict__ tok,
                                                         const float* __restrict__ emb,
                                                         float* __restrict__ x) {
  const int t0 = tok[0];
  int j = blockIdx.x * blockDim.x + threadIdx.x;
  if (j < GRU_H) x[j] = fmaxf(emb[(size_t)t0 * GRU_H + j], 0.0f);
}

// Computes gi = W_ih.x + b_ih and gh = W_hh.h + b_hh for one layer.
// Grid = 2*nb blocks of 256 threads; first nb blocks -> (W_ih, x, gi),
// last nb -> (W_hh, h, gh). Each block covers 32 rows: 8 waves = 2 row-groups
// of 16 x 4-way K split (K/4 = 256 per wave), reduced through LDS.
__global__ __launch_bounds__(256) void gru_gates_kernel(
    const float* __restrict__ Wih, const float* __restrict__ Whh,
    const float* __restrict__ bih, const float* __restrict__ bhh,
    const float* __restrict__ x, const float* __restrict__ h,
    float* __restrict__ gi, float* __restrict__ gh) {
  __shared__ float vs[GRU_H];
  __shared__ float part[8][16];
  const int R = 3 * GRU_H, K = GRU_H;

  const int nb = (int)gridDim.x >> 1;
  const int half = (blockIdx.x >= (unsigned)nb) ? 1 : 0;
  const int mb = (int)blockIdx.x - half * nb;
  const float* W = half ? Whh : Wih;
  const float* b = half ? bhh : bih;
  const float* vec = half ? h : x;
  float* y = half ? gh : gi;

  // Stage the 4 KB vector into LDS (256 threads x float4).
  ((float4*)vs)[threadIdx.x] = ((const float4*)vec)[threadIdx.x];
  __syncthreads();

  const int lane = (int)(threadIdx.x & 31u);
  const int wave = (int)(threadIdx.x >> 5);
  const int chunk = wave >> 2;      // which 16-row group (0/1)
  const int kq = wave & 3;          // K quarter
  const int rowBase = mb * 32 + chunk * 16;
  int row = rowBase + (lane & 15);
  if (row >= R) row = R - 1;        // clamp: keep EXEC all-1s for WMMA

  const int kchunk = K >> 2;
  v8f acc = wmma_row16_dot(W, K, vs, row, kq * kchunk, kq * kchunk + kchunk);

  if (lane == 0 || lane == 16) {
    const int off = (lane >> 4) * 8;
#pragma unroll
    for (int j = 0; j < 8; ++j) part[wave][off + j] = acc[j];
  }
  __syncthreads();

  if (threadIdx.x < 32) {
    const int t = (int)threadIdx.x;
    const int c = t >> 4, j = t & 15;
    const int wb = c * 4;
    float s = part[wb][j] + part[wb + 1][j] + part[wb + 2][j] + part[wb + 3][j];
    const int r = mb * 32 + c * 16 + j;
    if (r < R) y[r] = s + b[r];
  }
}

// r = sig(gi_r+gh_r); z = sig(gi_z+gh_z); n = tanh(gi_n + r*gh_n);
// h' = (1-z)*n + z*h.  Writes h' to both the next-layer input and d_out tail.
__global__ __launch_bounds__(256) void gru_combine_kernel(
    const float* __restrict__ gi, const float* __restrict__ gh,
    const float* __restrict__ hprev, float* __restrict__ xnext,
    float* __restrict__ hout) {
  int j = blockIdx.x * blockDim.x + threadIdx.x;
  if (j >= GRU_H) return;
  const int H = GRU_H;
  float r = 1.0f / (1.0f + expf(-(gi[j] + gh[j])));
  float z = 1.0f / (1.0f + expf(-(gi[j + H] + gh[j + H])));
  float n = tanhf(gi[j + 2 * H] + r * gh[j + 2 * H]);
  float hn = (1.0f - z) * n + z * hprev[j];
  xnext[j] = hn;
  hout[j] = hn;
}

// logits[v] = W_out[v] . x + b_out[v].  8 waves x 16 rows = 128 rows per block.
__global__ __launch_bounds__(256) void proj_kernel(const float* __restrict__ Wout,
                                                   const float* __restrict__ bout,
                                                   const float* __restrict__ x,
                                                   float* __restrict__ logits) {
  __shared__ float vs[GRU_H];
  const int R = GRU_V, K = GRU_H;
  ((float4*)vs)[threadIdx.x] = ((const float4*)x)[threadIdx.x];
  __syncthreads();

  const int lane = (int)(threadIdx.x & 31u);
  const int wave = (int)(threadIdx.x >> 5);
  const int rowBase = (int)blockIdx.x * 128 + wave * 16;
  int row = rowBase + (lane & 15);
  if (row >= R) row = R - 1;        // clamp for EXEC-all-1s WMMA

  v8f acc = wmma_row16_dot(Wout, K, vs, row, 0, K);

  if (lane == 0 || lane == 16) {
    const int base = rowBase + (lane >> 4) * 8;
#pragma unroll
    for (int j = 0; j < 8; ++j) {
      const int r = base + j;
      if (r < R) logits[r] = acc[j] + bout[r];
    }
  }
}

// Single block: logZ = max + log(sum(exp(l - max))) over V logits.
__global__ __launch_bounds__(256) void logsumexp_kernel(const float* __restrict__ logits,
                                                        float* __restrict__ logZ) {
  __shared__ float sm[256];
  const int t = (int)threadIdx.x;
  float mx = -INFINITY;
  for (int i = t; i < GRU_V; i += 256) mx = fmaxf(mx, logits[i]);
  sm[t] = mx;
  __syncthreads();
  for (int s = 128; s > 0; s >>= 1) {
    if (t < s) sm[t] = fmaxf(sm[t], sm[t + s]);
    __syncthreads();
  }
  const float gmax = sm[0];
  __syncthreads();
  float sum = 0.0f;
  for (int i = t; i < GRU_V; i += 256) sum += expf(logits[i] - gmax);
  sm[t] = sum;
  __syncthreads();
  for (int s = 128; s > 0; s >>= 1) {
    if (t < s) sm[t] += sm[t + s];
    __syncthreads();
  }
  if (t == 0) logZ[0] = gmax + logf(sm[0]);
}

__global__ __launch_bounds__(256) void finalize_kernel(float* __restrict__ logits,
                                                       const float* __restrict__ logZ) {
  int v = blockIdx.x * blockDim.x + threadIdx.x;
  if (v < GRU_V) logits[v] = logits[v] - logZ[0];
}

extern "C" void kernel_launch(void* const* d_in, const int* in_sizes, int n_in,
                              void* d_out, int out_size, void* d_ws, size_t ws_size,
                              hipStream_t stream) {
  (void)in_sizes; (void)n_in; (void)out_size; (void)ws_size;
  const int H = GRU_H, L = GRU_L, V = GRU_V;

  const int*   tok    = (const int*)d_in[0];
  const float* hidden = (const float*)d_in[1];   // (L,1,H)
  const float* emb    = (const float*)d_in[2];   // (V,H)
  const float* W_ih   = (const float*)d_in[3];   // (L,3H,H)
  const float* W_hh   = (const float*)d_in[4];   // (L,3H,H)
  const float* b_ih   = (const float*)d_in[5];   // (L,3H)
  const float* b_hh   = (const float*)d_in[6];   // (L,3H)
  const float* W_out  = (const float*)d_in[7];   // (V,H)
  const float* b_out  = (const float*)d_in[8];   // (V,)

  float* out = (float*)d_out;                    // [V logprobs][L*H hidden_new]
  float* ws  = (float*)d_ws;
  float* xbuf = ws;                              // H
  float* gi   = ws + H;                          // 3H
  float* gh   = ws + 4 * H;                      // 3H
  float* logZ = ws + 7 * H;                      // 1

  embed_relu_kernel<<<4, 256, 0, stream>>>(tok, emb, xbuf);

  for (int l = 0; l < L; ++l) {
    const size_t wofs = (size_t)l * 3 * H * H;
    // 96 blocks per matrix (32 rows each) x 2 matrices.
    gru_gates_kernel<<<192, 256, 0, stream>>>(W_ih + wofs, W_hh + wofs,
                                              b_ih + (size_t)l * 3 * H,
                                              b_hh + (size_t)l * 3 * H,
                                              xbuf, hidden + (size_t)l * H, gi, gh);
    gru_combine_kernel<<<4, 256, 0, stream>>>(gi, gh, hidden + (size_t)l * H,
                                              xbuf, out + V + (size_t)l * H);
  }

  proj_kernel<<<(V + 127) / 128, 256, 0, stream>>>(W_out, b_out, xbuf, out);
  logsumexp_kernel<<<1, 256, 0, stream>>>(out, logZ);
  finalize_kernel<<<(V + 255) / 256, 256, 0, stream>>>(out, logZ);
}